// CIntegration_11038065951535
// MI455X (gfx1250) — compile-verified
//
#include <hip/hip_runtime.h>
#include <hip/hip_bf16.h>
#include <stdint.h>

// Problem constants (match reference)
#define EMB      128
#define NTOTAL   128
#define LDS_STRIDE 129   // 128 dwords + 1 pad dword -> conflict-free column gathers

typedef unsigned int u32x4 __attribute__((ext_vector_type(4)));
typedef int          i32x4 __attribute__((ext_vector_type(4)));
typedef int          i32x8 __attribute__((ext_vector_type(8)));

__global__ __launch_bounds__(256) void onehot_linear_kernel(
    const float* __restrict__ vt,      // [NTOK, 128]
    const int*   __restrict__ rgap,    // [NTOK]
    const int*   __restrict__ sgap,    // [NTOK]
    const int*   __restrict__ pcount,  // [NTOK]
    const float* __restrict__ W,       // [128, 128] row-major (emb, ntotal)
    float*       __restrict__ out,     // [NTOK, 256]
    int ntok)
{
    // W staged in LDS with 1-dword pad per 128-dword row: W[e][n] at wlds[e*129 + n].
    __shared__ float wlds[EMB * LDS_STRIDE];

#if defined(__gfx1250__) && __has_builtin(__builtin_amdgcn_tensor_load_to_lds)
    // --- Tensor Data Mover: DMA W (64KB) global->LDS with padding, one wave issues ---
    if (threadIdx.x < 32) {
        uint64_t gaddr   = (uint64_t)(uintptr_t)W;
        uint32_t ldsaddr = (uint32_t)(uintptr_t)(&wlds[0]); // low 32 bits = LDS byte offset

        // D# group 0: count=1 (valid), lds_addr, global_addr[56:0], type=2 ("image")
        u32x4 g0 = {
            1u,
            ldsaddr,
            (uint32_t)(gaddr & 0xFFFFFFFFu),
            (uint32_t)((gaddr >> 32) & 0x01FFFFFFu) | 0x80000000u  // type=2 in bits[127:126]
        };
        // D# group 1:
        // d0: data_size=2 (4B)<<16 | pad_enable<<20 | pad_interval=6 (128 dwords)<<22 | pad_amount=0 (1 dword)
        // d1: tensor_dim0[15:0]<<16 = 128
        // d2: tensor_dim1[15:0]<<16 = 128
        // d3: tile_dim0<<16 = 128
        // d4: tile_dim1 = 128 (tile_dim2 = 0 -> 2D)
        // d5: tensor_dim0_stride = 128
        i32x8 g1 = { 0x01920000, 128 << 16, 128 << 16, 128 << 16, 128, 128, 0, 0 };
        i32x4 g2 = { 0, 0, 0, 0 };
        i32x4 g3 = { 0, 0, 0, 0 };

#if __clang_major__ >= 23
        i32x8 g1x = { 0, 0, 0, 0, 0, 0, 0, 0 };
        __builtin_amdgcn_tensor_load_to_lds(g0, g1, g2, g3, g1x, 0);
#else
        __builtin_amdgcn_tensor_load_to_lds(g0, g1, g2, g3, 0);
#endif
        __builtin_amdgcn_s_wait_tensorcnt(0);
    }
#else
    // Fallback: cooperative padded copy (coalesced global reads, conflict-free LDS writes)
    for (int i = threadIdx.x; i < EMB * NTOTAL; i += 256) {
        int e = i >> 7;
        int n = i & 127;
        wlds[e * LDS_STRIDE + n] = W[i];
    }
#endif
    __syncthreads();

    const int lane = threadIdx.x & 31;
    const int wave = threadIdx.x >> 5;
    const int wave_stride = gridDim.x * 8;

    for (int t = blockIdx.x * 8 + wave; t < ntok; t += wave_stride) {
        // Per-token indices (uniform across the wave; broadcast load)
        const int r = rgap[t];            // [0,32)
        const int s = 32 + sgap[t];       // [32,64)
        const int p = 64 + pcount[t];     // [64,128)

        const float* vtt = vt + (size_t)t * EMB;
        float* ot = out + (size_t)t * (EMB + NTOTAL);

#pragma unroll
        for (int k = 0; k < 4; ++k) {
            const int e = lane + 32 * k;           // strided -> conflict-free LDS banks
            const int base = e * LDS_STRIDE;
            // Cct[e] = W[e][r] + W[e][s] + W[e][p]  (column gather, bank-conflict-free)
            float c = wlds[base + r] + wlds[base + s] + wlds[base + p];
            // vt/out are pure streams (zero reuse): non-temporal keeps L2 resident for W.
            float v = __builtin_nontemporal_load(&vtt[e]);
            __builtin_nontemporal_store(v * c, &ot[e]);                  // theta
            // one-hot concat (r,s,p are in disjoint ranges -> at most one hit)
            float oh = (e == r || e == s || e == p) ? 1.0f : 0.0f;
            __builtin_nontemporal_store(oh, &ot[EMB + e]);
        }
    }
}

extern "C" void kernel_launch(void* const* d_in, const int* in_sizes, int n_in,
                              void* d_out, int out_size, void* d_ws, size_t ws_size,
                              hipStream_t stream) {
    (void)n_in; (void)out_size; (void)d_ws; (void)ws_size;

    const float* vt     = (const float*)d_in[0];
    const int*   rgap   = (const int*)d_in[1];
    const int*   sgap   = (const int*)d_in[2];
    const int*   pcount = (const int*)d_in[3];
    const float* W      = (const float*)d_in[4];
    float*       out    = (float*)d_out;

    const int ntok = in_sizes[1];  // B*S = 262144

    // 2048 blocks x 256 threads: 8 waves/block, 16 tokens per wave.
    // TDM W-staging cost: 2048 * 64KB of L2-resident reads (L2 = 192MB).
    dim3 grid(2048), block(256);
    onehot_linear_kernel<<<grid, block, 0, stream>>>(vt, rgap, sgap, pcount, W, out, ntok);
}